// Local2d_3942779977856
// MI455X (gfx1250) — compile-verified
//
#include <hip/hip_runtime.h>

typedef float v2f __attribute__((ext_vector_type(2)));
typedef float v8f __attribute__((ext_vector_type(8)));
typedef unsigned int v4u __attribute__((ext_vector_type(4)));
typedef int v4i __attribute__((ext_vector_type(4)));
typedef int v8i __attribute__((ext_vector_type(8)));

#define BATCH   64
#define CIN     64
#define COUT    128
#define HWDIM   1024          // 32*32 output pixels
#define KTOT    576           // CIN * 3 * 3
#define KC      64            // K chunk streamed through LDS
#define NCHUNK  9             // 576 / 64
#define LSTR    68            // KC + 4 pad; 68 % 64 == 4 -> conflict-free b64 frag reads

#if defined(__has_builtin)
#  if __has_builtin(__builtin_amdgcn_tensor_load_to_lds) && __has_builtin(__builtin_amdgcn_s_wait_tensorcnt)
#    define USE_TDM 1
#  endif
#endif
#ifndef USE_TDM
#  define USE_TDM 0
#endif

#if USE_TDM
// One TDM descriptor: 2D tile, 128 rows x 64 fp32, row stride HWDIM*KTOT elements,
// LDS padding 4 DWORDs after every 64 DWORDs -> LDS row stride = 68 floats (== LSTR).
__device__ __forceinline__ void tdm_load_weight_chunk(const float* weight, int hw, int k0,
                                                      float* ldsdst) {
    const unsigned long long ga =
        (unsigned long long)(uintptr_t)(weight + (size_t)hw * KTOT + (size_t)k0);
    const unsigned int lds = (unsigned int)(uintptr_t)ldsdst;

    v4u g0;
    g0.x = 1u;                                        // count=1 (valid), user mode
    g0.y = lds;                                       // lds_addr
    g0.z = (unsigned int)(ga & 0xffffffffu);          // global_addr[31:0]
    g0.w = (unsigned int)((ga >> 32) & 0x01ffffffu)   // global_addr[56:32]
         | (2u << 30);                                // type = 2 ("image")

    v8i g1;
    g1[0] = (2 << 16)        // data_size = 4 bytes
          | (1 << 20)        // pad_enable
          | (5 << 22)        // pad_interval: 64 DWORDs
          | (3 << 25);       // pad_amount:   4 DWORDs
    g1[1] = 0;                                   // atomic_barrier off; tensor_dim0[15:0]=0
    g1[2] = (int)(0x10u | (128u << 16));         // tensor_dim0=0x100000 (no OOB), tensor_dim1[15:0]=128
    g1[3] = (64 << 16);                          // tensor_dim1 hi=0, tile_dim0=64
    g1[4] = 128;                                 // tile_dim1=128, tile_dim2=0 (2D)
    g1[5] = (int)(HWDIM * KTOT);                 // tensor_dim0_stride = 589824 elements
    g1[6] = 0;                                   // stride hi / dim1_stride lo
    g1[7] = 0;                                   // dim1_stride hi (unused, 2D)

    const v4i g2 = {0, 0, 0, 0};
    const v4i g3 = {0, 0, 0, 0};
    const v8i g4 = {0, 0, 0, 0, 0, 0, 0, 0};     // extra group (clang-23 6-arg form), unused
    __builtin_amdgcn_tensor_load_to_lds(g0, g1, g2, g3, g4, 0);
}
#endif

// Fallback / non-TDM weight stage: coalesced float4 copy through VGPRs.
__device__ __forceinline__ void stage_weight_regs(const float* __restrict__ weight,
                                                  int hw, int k0, int t, float* sWbuf) {
    const int wr = t >> 4;            // row 0..15 (+16/pass)
    const int wq = (t & 15) * 4;      // float4 column
    #pragma unroll
    for (int p = 0; p < 8; ++p) {
        const int row = wr + p * 16;
        const float4 v = *(const float4*)(weight + ((long)row * HWDIM + hw) * KTOT + k0 + wq);
        *(float4*)(&sWbuf[row * LSTR + wq]) = v;
    }
}

__device__ __forceinline__ void gather_patches(const float* __restrict__ x,
                                               int h, int w, int k0, int t, float* sPbuf) {
    const int pb = t >> 2;            // batch row 0..63
    const int pq = (t & 3) * 16;      // k sub-block
    #pragma unroll 4
    for (int kk = 0; kk < 16; ++kk) {
        const int kg  = k0 + pq + kk;
        const int i   = kg / 9;
        const int rem = kg - i * 9;
        const int r   = rem / 3;
        const int cc  = rem - r * 3;
        const int hh  = h - 1 + r;
        const int ww  = w - 1 + cc;
        float val = 0.f;
        if ((unsigned)hh < 32u && (unsigned)ww < 32u)
            val = x[((pb * CIN + i) * 32 + hh) * 32 + ww];
        sPbuf[pb * LSTR + pq + kk] = val;
    }
}

__global__ __launch_bounds__(256)
void local2d_wmma_f32(const float* __restrict__ x,
                      const float* __restrict__ weight,
                      const float* __restrict__ bias,
                      float* __restrict__ out) {
    __shared__ float sP[2][BATCH * LSTR];   // patch chunks  [b][k]   2 x 17,408 B
    __shared__ float sW[2][COUT * LSTR];    // weight chunks [o][k]   2 x 34,816 B

    const int hw   = blockIdx.x;
    const int h    = hw >> 5;
    const int w    = hw & 31;
    const int t    = threadIdx.x;
    const int lane = t & 31;             // wave32
    const int wv   = t >> 5;             // wave 0..7
    const int mt   = wv >> 1;            // batch tile 0..3
    const int ntb  = (wv & 1) * 4;       // cout tile base: 0 or 4

    // Fragment addressing (ISA 7.12.2): lanes 0-15 hold K=k..k+1, lanes 16-31 K=k+2..k+3
    const int rowA  = mt * 16 + (lane & 15);
    const int colB0 = ntb * 16 + (lane & 15);
    const int koff  = (lane >> 4) * 2;

    v8f acc[4] = {v8f{}, v8f{}, v8f{}, v8f{}};

    // ---- prologue: stage chunk 0 into buffer 0 ----
#if USE_TDM
    if (wv == 0) tdm_load_weight_chunk(weight, hw, 0, &sW[0][0]);
#else
    stage_weight_regs(weight, hw, 0, t, &sW[0][0]);
#endif
    gather_patches(x, h, w, 0, t, &sP[0][0]);

    for (int c = 0; c < NCHUNK; ++c) {
        const int cur = c & 1;
        const int nxt = cur ^ 1;

        // ---- stage chunk c+1 (overlaps with compute of chunk c on TDM path) ----
        if (c + 1 < NCHUNK) {
#if USE_TDM
            if (wv == 0) tdm_load_weight_chunk(weight, hw, (c + 1) * KC, &sW[nxt][0]);
#else
            stage_weight_regs(weight, hw, (c + 1) * KC, t, &sW[nxt][0]);
#endif
            gather_patches(x, h, w, (c + 1) * KC, t, &sP[nxt][0]);
        }

#if USE_TDM
        // Oldest outstanding TDM (chunk c) must be done; keep chunk c+1 in flight.
        if (wv == 0) {
            if (c + 1 < NCHUNK) __builtin_amdgcn_s_wait_tensorcnt(1);
            else                __builtin_amdgcn_s_wait_tensorcnt(0);
        }
#endif
        __syncthreads();   // publish chunk c (TDM completion + gather stores)

        // ---- compute: 16 K-steps of V_WMMA_F32_16X16X4_F32 ----
        const float* __restrict__ P = &sP[cur][0];
        const float* __restrict__ W = &sW[cur][0];
        #pragma unroll
        for (int kb = 0; kb < KC; kb += 4) {
            const int kidx = kb + koff;
            const v2f a  = *(const v2f*)(&P[rowA * LSTR + kidx]);
            const v2f b0 = *(const v2f*)(&W[(colB0 +  0) * LSTR + kidx]);
            const v2f b1 = *(const v2f*)(&W[(colB0 + 16) * LSTR + kidx]);
            const v2f b2 = *(const v2f*)(&W[(colB0 + 32) * LSTR + kidx]);
            const v2f b3 = *(const v2f*)(&W[(colB0 + 48) * LSTR + kidx]);
            acc[0] = __builtin_amdgcn_wmma_f32_16x16x4_f32(false, a, false, b0, (short)0, acc[0], false, false);
            acc[1] = __builtin_amdgcn_wmma_f32_16x16x4_f32(false, a, false, b1, (short)0, acc[1], false, false);
            acc[2] = __builtin_amdgcn_wmma_f32_16x16x4_f32(false, a, false, b2, (short)0, acc[2], false, false);
            acc[3] = __builtin_amdgcn_wmma_f32_16x16x4_f32(false, a, false, b3, (short)0, acc[3], false, false);
        }
        __syncthreads();   // all reads of 'cur' done before it is overwritten next iter
    }

    // ---- epilogue: bias add + store (C/D layout: VGPR j -> M=j (+8 hi half), N=lane&15) ----
    const int nloc  = lane & 15;
    const int mhalf = (lane >> 4) * 8;
    #pragma unroll
    for (int nt = 0; nt < 4; ++nt) {
        const int o  = (ntb + nt) * 16 + nloc;
        const float bv = bias[o * HWDIM + hw];
        #pragma unroll
        for (int j = 0; j < 8; ++j) {
            const int bm = mt * 16 + mhalf + j;
            out[((long)bm * COUT + o) * HWDIM + hw] = acc[nt][j] + bv;
        }
    }
}

extern "C" void kernel_launch(void* const* d_in, const int* in_sizes, int n_in,
                              void* d_out, int out_size, void* d_ws, size_t ws_size,
                              hipStream_t stream) {
    const float* x      = (const float*)d_in[0];
    const float* weight = (const float*)d_in[1];
    const float* bias   = (const float*)d_in[2];
    float* out          = (float*)d_out;
    (void)in_sizes; (void)n_in; (void)out_size; (void)d_ws; (void)ws_size;

    local2d_wmma_f32<<<HWDIM, 256, 0, stream>>>(x, weight, bias, out);
}